// ResGCN_64665027609332
// MI455X (gfx1250) — compile-verified
//
#include <hip/hip_runtime.h>

// ---------------------------------------------------------------------------
// ResGCN for MI455X (gfx1250).
//  - Dense GEMMs: V_WMMA_F32_16X16X4_F32, weights pre-repacked into fragment
//    order and staged to LDS via GLOBAL_LOAD_ASYNC_TO_LDS_B128 (ASYNCcnt).
//  - Segment-sums: global_atomic_add_f32 into an L2-resident (51.2MB) accum.
// ---------------------------------------------------------------------------

typedef float v2f __attribute__((ext_vector_type(2)));
typedef float v8f __attribute__((ext_vector_type(8)));

// ---------------- degree / weight prep ----------------

__global__ __launch_bounds__(256) void k_fill1(float* __restrict__ p, int n) {
  int i = blockIdx.x * 256 + threadIdx.x;
  if (i < n) p[i] = 1.0f;   // self-loop contributes 1 to every degree
}

__global__ __launch_bounds__(256) void k_degcount(const int* __restrict__ dst,
                                                  float* __restrict__ deg, int E) {
  int i = blockIdx.x * 256 + threadIdx.x;
  if (i < E) unsafeAtomicAdd(&deg[dst[i]], 1.0f);
}

__global__ __launch_bounds__(256) void k_rsqrt(const float* __restrict__ deg,
                                               float* __restrict__ dinv, int n) {
  int i = blockIdx.x * 256 + threadIdx.x;
  if (i < n) dinv[i] = rsqrtf(deg[i]);
}

// ---------------- weight repack into WMMA B-fragment order ----------------
// Bf[f], f = (k4*NTG + t2)*32 + lane, float4 = { b.x, b.y (tile 2*t2),
//                                                b.x, b.y (tile 2*t2+1) }
// where for k-step k=4*k4:  b.x = W[k+2*half][col], b.y = W[k+2*half+1][col],
// col = tile*16 + (lane&15), half = lane>>4.  Columns >= ncols_valid -> 0.

template <int NCOLS>
__global__ __launch_bounds__(256) void k_repack(const float* __restrict__ W, int ldw,
                                                int ncols_valid,
                                                float4* __restrict__ Bf) {
  constexpr int NT  = NCOLS / 16;
  constexpr int NTG = (NT + 1) / 2;
  int f = blockIdx.x * 256 + threadIdx.x;
  if (f >= 32 * NTG * 32) return;
  int lane = f & 31;
  int t2   = (f >> 5) % NTG;
  int k4   = (f >> 5) / NTG;
  int half = lane >> 4, l16 = lane & 15;
  int r0 = 4 * k4 + 2 * half;
  int c0 = (2 * t2) * 16 + l16;
  int c1 = c0 + 16;
  bool v1 = (2 * t2 + 1 < NT) && (c1 < ncols_valid);
  float4 o;
  o.x = (c0 < ncols_valid) ? W[(size_t)r0 * ldw + c0] : 0.f;
  o.y = (c0 < ncols_valid) ? W[(size_t)(r0 + 1) * ldw + c0] : 0.f;
  o.z = v1 ? W[(size_t)r0 * ldw + c1] : 0.f;
  o.w = v1 ? W[(size_t)(r0 + 1) * ldw + c1] : 0.f;
  Bf[f] = o;
}

// ---------------- WMMA GEMM: C[n_rows x NCOLS] = A[n_rows x 128] * B (+bias) ----
// Block = 256 threads = 8 waves; each wave owns a 16-row x NCOLS strip.
// B fragments are staged to LDS with async-to-LDS loads, then read as b128.

template <int NCOLS>
__global__ __launch_bounds__(256) void k_gemm_wmma(const float* __restrict__ A,
                                                   const float4* __restrict__ Bf,
                                                   const float* __restrict__ bias,
                                                   float* __restrict__ C,
                                                   int n_rows, int ldc) {
  constexpr int NT  = NCOLS / 16;
  constexpr int NTG = (NT + 1) / 2;
  constexpr int TOT = 32 * NTG * 32;          // float4 elements (4096 -> 64KB max)
  __shared__ float4 Bs[TOT];

  const int tid = threadIdx.x;

  // ---- async stage of repacked weights into LDS (CDNA5 ASYNCcnt path) ----
  for (int i = tid; i < TOT; i += 256) {
    unsigned lds_addr  = (unsigned)(size_t)(&Bs[i]);
    unsigned long long gaddr = (unsigned long long)(size_t)(Bf + i);
    asm volatile("global_load_async_to_lds_b128 %0, %1, off"
                 :: "v"(lds_addr), "v"(gaddr) : "memory");
  }
  asm volatile("s_wait_asynccnt 0" ::: "memory");
  __syncthreads();

  const int wave = tid >> 5;
  const int lane = tid & 31;
  const int half = lane >> 4;
  const int l16  = lane & 15;

  // Clamp A row instead of predicating the load: keeps EXEC all-ones (WMMA
  // requirement) and avoids per-iteration exec save/restore. Clamped rows only
  // feed D rows that are never stored.
  int arow = blockIdx.x * 128 + wave * 16 + l16;
  int ar   = arow < n_rows ? arow : (n_rows - 1);
  const float* __restrict__ ap = A + (size_t)ar * 128 + 2 * half;

  v8f acc[NT];
  const v8f vzero = {0.f, 0.f, 0.f, 0.f, 0.f, 0.f, 0.f, 0.f};
#pragma unroll
  for (int t = 0; t < NT; ++t) acc[t] = vzero;

#pragma unroll 4
  for (int k4 = 0; k4 < 32; ++k4) {
    v2f a = *(const v2f*)(ap + 4 * k4);   // A 16x4 frag: lanes0-15 K=k..k+1, 16-31 K=k+2..k+3
#pragma unroll
    for (int t2 = 0; t2 < NTG; ++t2) {
      float4 bb = Bs[(k4 * NTG + t2) * 32 + lane];   // 512B contiguous per wave
      v2f b0 = {bb.x, bb.y};
      acc[2 * t2] = __builtin_amdgcn_wmma_f32_16x16x4_f32(
          false, a, false, b0, (short)0, acc[2 * t2], false, false);
      if (2 * t2 + 1 < NT) {
        v2f b1 = {bb.z, bb.w};
        acc[2 * t2 + 1] = __builtin_amdgcn_wmma_f32_16x16x4_f32(
            false, a, false, b1, (short)0, acc[2 * t2 + 1], false, false);
      }
    }
  }

#pragma unroll
  for (int t = 0; t < NT; ++t) {
    const int col = t * 16 + l16;
    const float bv = bias ? bias[col] : 0.0f;
#pragma unroll
    for (int r = 0; r < 8; ++r) {
      const int orow = blockIdx.x * 128 + wave * 16 + r + 8 * half;
      if (orow < n_rows) C[(size_t)orow * ldc + col] = acc[t][r] + bv;
    }
  }
}

// ---------------- self-loop init (also fully initializes accumulator) ----------------

__global__ __launch_bounds__(256) void k_selfinit128(const float* __restrict__ S,
                                                     const float* __restrict__ dinv,
                                                     float* __restrict__ O, int n) {
  size_t gid = (size_t)blockIdx.x * 256 + threadIdx.x;
  if (gid >= (size_t)n * 128) return;
  int row = (int)(gid >> 7);
  float di = dinv[row];
  O[gid] = S[gid] * di * di;   // w_self = 1/deg = dinv^2
}

__global__ __launch_bounds__(256) void k_selfinit_gen(const float* __restrict__ S, int lds_,
                                                      const float* __restrict__ dinv,
                                                      float* __restrict__ O, int ldo,
                                                      int ncols, int n) {
  size_t gid = (size_t)blockIdx.x * 256 + threadIdx.x;
  if (gid >= (size_t)n * ncols) return;
  int row = (int)(gid / ncols);
  int col = (int)(gid % ncols);
  float di = dinv[row];
  O[(size_t)row * ldo + col] = S[(size_t)row * lds_ + col] * di * di;
}

// ---------------- edge scatter: O[dst] += S[src] * dinv[src]*dinv[dst] ----------------

__global__ __launch_bounds__(256) void k_scatter128(const int* __restrict__ src,
                                                    const int* __restrict__ dst,
                                                    const float* __restrict__ dinv,
                                                    const float* __restrict__ S,
                                                    float* __restrict__ O, int E) {
  int gid  = blockIdx.x * 256 + threadIdx.x;
  int e    = gid >> 5;           // one wave32 per edge
  int lane = gid & 31;           // lane covers 4 consecutive floats of the row
  if (e >= E) return;
  int s = src[e], d = dst[e];
  float w = dinv[s] * dinv[d];
  const float4 v = *(const float4*)(S + (size_t)s * 128 + lane * 4);
  float* o = O + (size_t)d * 128 + lane * 4;
  unsafeAtomicAdd(o + 0, v.x * w);
  unsafeAtomicAdd(o + 1, v.y * w);
  unsafeAtomicAdd(o + 2, v.z * w);
  unsafeAtomicAdd(o + 3, v.w * w);
}

__global__ __launch_bounds__(256) void k_scatter_gen(const int* __restrict__ src,
                                                     const int* __restrict__ dst,
                                                     const float* __restrict__ dinv,
                                                     const float* __restrict__ S, int lds_,
                                                     float* __restrict__ O, int ldo,
                                                     int ncols, int E) {
  int gid  = blockIdx.x * 256 + threadIdx.x;
  int e    = gid >> 5;
  int lane = gid & 31;
  if (e >= E) return;
  int s = src[e], d = dst[e];
  float w = dinv[s] * dinv[d];
  for (int c = lane; c < ncols; c += 32)
    unsafeAtomicAdd(O + (size_t)d * ldo + c, S[(size_t)s * lds_ + c] * w);
}

// ---------------- epilogue: h = relu(O) + x0 (in-place safe) ----------------

__global__ __launch_bounds__(256) void k_relu_res(const float* __restrict__ O,
                                                  const float* __restrict__ x0,
                                                  float* __restrict__ h, int n) {
  size_t gid = (size_t)blockIdx.x * 256 + threadIdx.x;
  if (gid >= (size_t)n * 128) return;
  h[gid] = fmaxf(O[gid], 0.0f) + x0[gid];
}

// ---------------------------------------------------------------------------

extern "C" void kernel_launch(void* const* d_in, const int* in_sizes, int n_in,
                              void* d_out, int out_size, void* d_ws, size_t ws_size,
                              hipStream_t stream) {
  const float* x  = (const float*)d_in[0];
  const int*   src = (const int*)d_in[1];
  const int*   dst = (const int*)d_in[2];
  const float* Wt = (const float*)d_in[3];
  const float* bt = (const float*)d_in[4];
  const float* W0 = (const float*)d_in[5];
  const float* W1 = (const float*)d_in[6];
  const float* W2 = (const float*)d_in[7];
  const float* W3 = (const float*)d_in[8];
  float* out = (float*)d_out;

  const int n = in_sizes[0] / 128;
  const int E = in_sizes[1];

  char* p = (char*)d_ws;
  auto carve = [&](size_t bytes) -> char* {
    char* r = p;
    p += (bytes + 255) & ~(size_t)255;
    return r;
  };
  float*  deg  = (float*)carve((size_t)n * 4);
  float*  dinv = (float*)carve((size_t)n * 4);
  float*  x0   = (float*)carve((size_t)n * 128 * 4);
  float*  bufA = (float*)carve((size_t)n * 128 * 4);   // support S
  float*  bufB = (float*)carve((size_t)n * 128 * 4);   // accum / h
  float*  bufC = (float*)carve((size_t)n * 128 * 4);   // accum / h
  float4* WtR  = (float4*)carve(32 * 4 * 32 * 16);     // 64KB fragment-packed
  float4* W0R  = (float4*)carve(32 * 4 * 32 * 16);
  float4* W1R  = (float4*)carve(32 * 4 * 32 * 16);
  float4* W2R  = (float4*)carve(32 * 4 * 32 * 16);
  float4* W3R  = (float4*)carve(32 * 2 * 32 * 16);     // 32KB (48-col padded)

  const dim3 b256(256);
  const int gN    = (n + 255) / 256;
  const int gE    = (E + 255) / 256;
  const int gRow  = (n + 127) / 128;
  const int gNE   = (int)(((size_t)n * 128 + 255) / 256);
  const int gScat = (int)(((size_t)E * 32 + 255) / 256);
  const int gOut  = (int)(((size_t)n * 40 + 255) / 256);
  const int gRp8  = (32 * 4 * 32 + 255) / 256;   // repack grid, NT=8
  const int gRp3  = (32 * 2 * 32 + 255) / 256;   // repack grid, NT=3

  // normalization: deg = 1 + indeg; dinv = rsqrt(deg)
  k_fill1<<<gN, b256, 0, stream>>>(deg, n);
  k_degcount<<<gE, b256, 0, stream>>>(dst, deg, E);
  k_rsqrt<<<gN, b256, 0, stream>>>(deg, dinv, n);

  // repack all weights into WMMA fragment order (W3 zero-padded 40 -> 48)
  k_repack<128><<<gRp8, b256, 0, stream>>>(Wt, 128, 128, WtR);
  k_repack<128><<<gRp8, b256, 0, stream>>>(W0, 128, 128, W0R);
  k_repack<128><<<gRp8, b256, 0, stream>>>(W1, 128, 128, W1R);
  k_repack<128><<<gRp8, b256, 0, stream>>>(W2, 128, 128, W2R);
  k_repack<48><<<gRp3, b256, 0, stream>>>(W3, 40, 40, W3R);

  // residual anchor x0 = x @ Wt + bt
  k_gemm_wmma<128><<<gRow, b256, 0, stream>>>(x, WtR, bt, x0, n, 128);

  // layer 0: h(bufB) = relu(conv(x, W0)) + x0
  k_gemm_wmma<128><<<gRow, b256, 0, stream>>>(x, W0R, nullptr, bufA, n, 128);
  k_selfinit128<<<gNE, b256, 0, stream>>>(bufA, dinv, bufB, n);
  k_scatter128<<<gScat, b256, 0, stream>>>(src, dst, dinv, bufA, bufB, E);
  k_relu_res<<<gNE, b256, 0, stream>>>(bufB, x0, bufB, n);

  // layer 1: h(bufC) = relu(conv(bufB, W1)) + x0
  k_gemm_wmma<128><<<gRow, b256, 0, stream>>>(bufB, W1R, nullptr, bufA, n, 128);
  k_selfinit128<<<gNE, b256, 0, stream>>>(bufA, dinv, bufC, n);
  k_scatter128<<<gScat, b256, 0, stream>>>(src, dst, dinv, bufA, bufC, E);
  k_relu_res<<<gNE, b256, 0, stream>>>(bufC, x0, bufC, n);

  // layer 2: h(bufB) = relu(conv(bufC, W2)) + x0
  k_gemm_wmma<128><<<gRow, b256, 0, stream>>>(bufC, W2R, nullptr, bufA, n, 128);
  k_selfinit128<<<gNE, b256, 0, stream>>>(bufA, dinv, bufB, n);
  k_scatter128<<<gScat, b256, 0, stream>>>(src, dst, dinv, bufA, bufB, E);
  k_relu_res<<<gNE, b256, 0, stream>>>(bufB, x0, bufB, n);

  // output: logits = conv(bufB, W3)  (128 -> 40, padded to 48 for WMMA tiles)
  k_gemm_wmma<48><<<gRow, b256, 0, stream>>>(bufB, W3R, nullptr, bufA, n, 48);
  k_selfinit_gen<<<gOut, b256, 0, stream>>>(bufA, 48, dinv, out, 40, 40, n);
  k_scatter_gen<<<gScat, b256, 0, stream>>>(src, dst, dinv, bufA, 48, out, 40, 40, E);
}